// DS3_Dempster_33612414059094
// MI455X (gfx1250) — compile-verified
//
#include <hip/hip_runtime.h>
#include <stdint.h>

// ---------------------------------------------------------------------------
// Dempster-Shafer sequential scan, B=4096 rows, D=128 steps, C=101 channels.
// Memory-bound (211.8 MB @ 23.3 TB/s ~= 9.1 us). Strategy:
//   * one wave32 per row; lanes 0..24 hold 4 contiguous channels (float4),
//     lane 25 holds omega, lanes 26..31 hold zeros
//   * combine is scale-invariant in m1 -> normalize only every 8 steps
//   * stream D in 16 tiles of 8 steps via gfx1250 async global->LDS copies
//     (ASYNCcnt), triple-buffered, depth-2 prefetch, s_wait_asynccnt gating
//   * LDS row stride 512 B with pre-zeroed pad -> unpredicated ds_load_b128,
//     lane 25 reads (omega,0,0,0) for free
// ---------------------------------------------------------------------------

#define WAIT_ASYNC(N) asm volatile("s_wait_asynccnt " N ::: "memory")

__device__ __forceinline__ void async_b128(uint32_t lds_off, uint32_t voff,
                                           const void* sbase) {
  // GVS mode: mem_addr = SGPR64 + VGPR32_offset ; LDS dest byte addr in VGPR
  asm volatile("global_load_async_to_lds_b128 %0, %1, %2"
               :: "v"(lds_off), "v"(voff), "s"(sbase) : "memory");
}
__device__ __forceinline__ void async_b32(uint32_t lds_off, uint32_t voff,
                                          const void* sbase) {
  asm volatile("global_load_async_to_lds_b32 %0, %1, %2"
               :: "v"(lds_off), "v"(voff), "s"(sbase) : "memory");
}

__global__ __launch_bounds__(128) void ds_scan_kernel(
    const float* __restrict__ in, float* __restrict__ out) {
  constexpr int C = 101;
  constexpr int ROW_B = 128 * 101 * 4;   // 51712 bytes per batch row
  constexpr int STEP_B = 101 * 4;        // 404 bytes per D-step
  constexpr int TILE_B = 8 * STEP_B;     // 3232 bytes per 8-step tile
  constexpr int LROWF = 128;             // LDS row stride: 128 floats (512 B)

  // 3 buffers * 4 waves * 8 steps * 128 floats = 12288 floats = 48 KB
  __shared__ __align__(16) float lds[3 * 4 * 8 * LROWF];

  const int lane = threadIdx.x & 31;
  const int wv   = threadIdx.x >> 5;
  const int b    = blockIdx.x * 4 + wv;

  // ---- zero the pad region (floats 101..127 of each of this wave's 24 rows)
  for (int i = lane; i < 24 * 27; i += 32) {
    int row = i / 27, p = i - row * 27;         // row in [0,24): buf*8 + step
    int buf = row >> 3, s = row & 7;
    lds[((buf * 4 + wv) * 8 + s) * LROWF + 101 + p] = 0.0f;
  }

  // ---- per-lane async copy offsets (constant across tiles)
  // tile = 8 steps * (25 b128 chunks + 1 b32 omega); flatten b128 chunks:
  // chunk q = 32*j + lane, q < 200 ; step s = q/25, chunk-in-step cc = q%25
  uint32_t gof[7], lof[7];
#pragma unroll
  for (int j = 0; j < 7; ++j) {
    int q = 32 * j + lane;
    int s = q / 25;
    int cc = q - 25 * s;
    gof[j] = (uint32_t)(s * STEP_B + 16 * cc);
    lof[j] = (uint32_t)(s * (LROWF * 4) + 16 * cc);
  }
  const uint32_t gofw = (uint32_t)(lane * STEP_B + 400);   // omega, lanes 0..7
  const uint32_t lofw = (uint32_t)(lane * (LROWF * 4) + 400);

  const uint32_t grow = (uint32_t)b * (uint32_t)ROW_B;

  auto issue = [&](int t) {
    const int bi = t % 3;
    const uint32_t g  = grow + (uint32_t)t * (uint32_t)TILE_B;
    const uint32_t lb = (uint32_t)((bi * 4 + wv) * (8 * LROWF * 4));
#pragma unroll
    for (int j = 0; j < 6; ++j)                 // 192 full chunks, all lanes
      async_b128(lb + lof[j], g + gof[j], (const void*)in);
    if (lane < 8) {                             // tail chunks + 8 omegas
      async_b128(lb + lof[6], g + gof[6], (const void*)in);
      async_b32(lb + lofw, g + gofw, (const void*)in);
    }
  };

  issue(0);
  issue(1);

  float m0 = 0.f, m1 = 0.f, m2c = 0.f, m3 = 0.f;  // this lane's 4 channels
  float w1 = 0.f;                                  // replicated omega mass

  for (int t = 0; t < 16; ++t) {
    if (t + 2 < 16) issue(t + 2);
    // each tile = 8 async instrs; <=16 outstanding => oldest tile landed
    if (t < 14)       { WAIT_ASYNC("16"); }
    else if (t == 14) { WAIT_ASYNC("8"); }
    else              { WAIT_ASYNC("0"); }

    const int bi = t % 3;
    const int base_f = (bi * 4 + wv) * 8 * LROWF;

    int s0 = 0;
    if (t == 0) {  // m_init = inputs[:,0,:]
      const float4 v = ((const float4*)(lds + base_f))[lane];
      m0 = v.x; m1 = v.y; m2c = v.z; m3 = v.w;
      w1 = lds[base_f + 100];
      s0 = 1;
    }
#pragma unroll
    for (int s = 0; s < 8; ++s) {
      if (s < s0) continue;
      const int rf = base_f + s * LROWF;
      const float4 v = ((const float4*)(lds + rf))[lane];  // ds_load_b128
      const float w2 = lds[rf + 100];                      // broadcast b32
      // m <- m*(m2 + w2) + w1*m2   (unnormalized combine; scale-invariant)
      const float u0 = w1 * v.x, u1 = w1 * v.y, u2 = w1 * v.z, u3 = w1 * v.w;
      m0  = fmaf(m0,  v.x + w2, u0);
      m1  = fmaf(m1,  v.y + w2, u1);
      m2c = fmaf(m2c, v.z + w2, u2);
      m3  = fmaf(m3,  v.w + w2, u3);
      w1  = fmaf(w1,  w2 + w2,  w1 * w2);  // bit-identical to lane25's m0
    }

    // rescale every 8 steps (keeps range safe; final pass normalizes)
    float p = (m0 + m1) + (m2c + m3);
#pragma unroll
    for (int o = 16; o >= 1; o >>= 1) p += __shfl_xor(p, o, 32);
    const float invp = 1.0f / p;
    m0 *= invp; m1 *= invp; m2c *= invp; m3 *= invp; w1 *= invp;
  }

  // ---- store (rows are only 4B aligned -> b32 stores)
  const float r[4] = {m0, m1, m2c, m3};
#pragma unroll
  for (int i = 0; i < 4; ++i) {
    const int c = 4 * lane + i;
    if (c < C) out[(size_t)b * C + c] = r[i];
  }
}

extern "C" void kernel_launch(void* const* d_in, const int* in_sizes, int n_in,
                              void* d_out, int out_size, void* d_ws,
                              size_t ws_size, hipStream_t stream) {
  const float* in = (const float*)d_in[0];
  float* out = (float*)d_out;
  (void)in_sizes; (void)n_in; (void)out_size; (void)d_ws; (void)ws_size;
  dim3 grid(4096 / 4), block(128);
  ds_scan_kernel<<<grid, block, 0, stream>>>(in, out);
}